// Complex_Only_46308337385506
// MI455X (gfx1250) — compile-verified
//
#include <hip/hip_runtime.h>
#include <hip/hip_bf16.h>

typedef float v2f __attribute__((ext_vector_type(2)));
typedef float v8f __attribute__((ext_vector_type(8)));
typedef int   v4i __attribute__((ext_vector_type(4)));
typedef __attribute__((address_space(1))) v4i as1_v4i;
typedef __attribute__((address_space(3))) v4i as3_v4i;
typedef __attribute__((address_space(3))) void as3_void;

#define EPSF 1e-6f
#define NEGS 0.2f

// problem sizes
constexpr int kB = 8, kC = 16384, kE = 64, kF = 64;
constexpr long kNS = (long)kB * kC;        // 131072 sites
constexpr int  kTS = 16;                   // sites per tile (WMMA N)
constexpr int  kNT = (int)(kNS / kTS);     // 8192 tiles
constexpr int  PAD = 66;                   // row pad (floats) -> bank-conflict free b64 reads

// LDS layout (in floats)
constexpr int W_SZ   = 64 * PAD;           // one 64x64 matrix, padded
constexpr int W_OFF  = 0;                  // Wa,Wb,Wc,Wd
constexpr int T_SZ   = kTS * PAD;          // one (term,i) block: [site][e]
constexpr int T_OFF  = W_OFF + 4 * W_SZ;   // 9 blocks: term t in {a,b,c} x i in {0,1,2}
constexpr int Y_OFF  = T_OFF + 9 * T_SZ;   // 3 blocks: Y staged as [i][site][g]
constexpr int XJ_SZ  = kTS * kE * 3;       // 3072 floats: one tile of X or J (contiguous!)
constexpr int XJ_OFF = Y_OFF + 3 * T_SZ;   // double buffer: [2][X(3072) | J(3072)]
constexpr int LDS_FLOATS = XJ_OFF + 2 * 2 * XJ_SZ;   // 41856 floats = 167,424 B

#define TERM_IDX(t, i) (T_OFF + ((t) * 3 + (i)) * T_SZ)

__device__ __forceinline__ void async_tile_prefetch(
    const float* __restrict__ X, const float* __restrict__ J,
    float* lds, int buf, long site0, int tid)
{
    // one tile of X and J each = 16 sites * 64 e * 3 comp * 4B = 12,288 B, contiguous.
    // 128 threads * 16B = 2 KB per block-wide async op -> 6 ops per tensor.
    const float* gx = X + site0 * (kE * 3);
    const float* gj = J + site0 * (kE * 3);
    float* lx = lds + XJ_OFF + buf * (2 * XJ_SZ);
    float* lj = lx + XJ_SZ;
    #pragma unroll
    for (int q = 0; q < 6; ++q) {
        int off = (tid + q * 128) * 4;     // float index, 16B per lane
#if __has_builtin(__builtin_amdgcn_global_load_async_to_lds_b128)
        __builtin_amdgcn_global_load_async_to_lds_b128(
            (as1_v4i*)(gx + off), (as3_v4i*)(lx + off), 0, 0);
        __builtin_amdgcn_global_load_async_to_lds_b128(
            (as1_v4i*)(gj + off), (as3_v4i*)(lj + off), 0, 0);
#else
        unsigned lxa = (unsigned)(size_t)(as3_void*)(lx + off);
        unsigned lja = (unsigned)(size_t)(as3_void*)(lj + off);
        asm volatile("global_load_async_to_lds_b128 %0, %1, off"
                     :: "v"(lxa), "v"(gx + off) : "memory");
        asm volatile("global_load_async_to_lds_b128 %0, %1, off"
                     :: "v"(lja), "v"(gj + off) : "memory");
#endif
    }
}

__launch_bounds__(128)
extern "C" __global__ void vn_complex_fused_kernel(
    const float* __restrict__ X, const float* __restrict__ J,
    const float* __restrict__ Wa, const float* __restrict__ Wb,
    const float* __restrict__ Wc, const float* __restrict__ Wd,
    float* __restrict__ out)
{
    extern __shared__ float lds[];
    const int tid   = threadIdx.x;
    const int lane  = tid & 31;
    const int wave  = tid >> 5;        // 4 waves
    const int lhalf = lane >> 4;       // 0/1 : which 16-lane half
    const int lmod  = lane & 15;       // N (column) within tile / M row within frag

    // ---- load the four 64x64 weight matrices into padded LDS (once) ----
    for (int idx = tid; idx < 4 * 64 * 64; idx += 128) {
        int m = idx >> 12;
        int r = (idx >> 6) & 63;
        int c = idx & 63;
        const float* Wm = (m == 0) ? Wa : (m == 1) ? Wb : (m == 2) ? Wc : Wd;
        lds[W_OFF + m * W_SZ + r * PAD + c] = Wm[r * 64 + c];
    }

    // ---- prologue: async-prefetch first tile's X/J into buffer 0 ----
    int buf = 0;
    if (blockIdx.x < kNT)
        async_tile_prefetch(X, J, lds, buf, (long)blockIdx.x * kTS, tid);
    asm volatile("s_wait_asynccnt 0x0" ::: "memory");
    __syncthreads();                                  // also publishes W

    const int arow = wave * 16 + lmod;     // A-fragment row (f) for this lane

    for (int tile = blockIdx.x; tile < kNT; tile += gridDim.x) {
        const long site0 = (long)tile * kTS;

        // ================= front end: basis + a/b/c terms (from LDS) =================
        const float* xb = lds + XJ_OFF + buf * (2 * XJ_SZ);   // [s*192 + e*3 + c]
        const float* jb = xb + XJ_SZ;
        for (int p = tid; p < kTS * kE; p += 128) {
            int e = p & 63;
            int s = p >> 6;
            int o3 = s * (kE * 3) + e * 3;
            float jx = jb[o3 + 0], jy = jb[o3 + 1], jz = jb[o3 + 2];
            float x0 = xb[o3 + 0], x1 = xb[o3 + 1], x2 = xb[o3 + 2];

            float inj = 1.f / (sqrtf(jx * jx + jy * jy + jz * jz) + EPSF);
            float nJ0 = jx * inj, nJ1 = jy * inj, nJ2 = jz * inj;
            float ss  = nJ0 * nJ0 + nJ1 * nJ1;
            float uz  = -ss / (nJ2 + EPSF);
            float inu = 1.f / (sqrtf(ss + uz * uz) + EPSF);
            float nU0 = nJ0 * inu, nU1 = nJ1 * inu, nU2 = uz * inu;
            // nV = cross(nU, nJ)
            float nV0 = nU1 * nJ2 - nU2 * nJ1;
            float nV1 = nU2 * nJ0 - nU0 * nJ2;
            float nV2 = nU0 * nJ1 - nU1 * nJ0;
            // RTX_i = nU_i*x0 + nV_i*x1 + nJ_i*x2
            float rt0 = nU0 * x0 + nV0 * x1 + nJ0 * x2;
            float rt1 = nU1 * x0 + nV1 * x1 + nJ1 * x2;
            float rt2 = nU2 * x0 + nV2 * x1 + nJ2 * x2;
            // a[j] = R[j,0]*rt0 + R[j,1]*rt1 ; b[j] = R[j,1]*rt0 - R[j,0]*rt1 ; c[j] = R[j,2]*rt2
            int o = s * PAD + e;
            lds[TERM_IDX(0, 0) + o] = nU0 * rt0 + nU1 * rt1;
            lds[TERM_IDX(0, 1) + o] = nV0 * rt0 + nV1 * rt1;
            lds[TERM_IDX(0, 2) + o] = nJ0 * rt0 + nJ1 * rt1;
            lds[TERM_IDX(1, 0) + o] = nU1 * rt0 - nU0 * rt1;
            lds[TERM_IDX(1, 1) + o] = nV1 * rt0 - nV0 * rt1;
            lds[TERM_IDX(1, 2) + o] = nJ1 * rt0 - nJ0 * rt1;
            lds[TERM_IDX(2, 0) + o] = nU2 * rt2;
            lds[TERM_IDX(2, 1) + o] = nV2 * rt2;
            lds[TERM_IDX(2, 2) + o] = nJ2 * rt2;
        }
        __syncthreads();                               // barrier 1: terms ready

        // kick off async prefetch of the NEXT tile while the GEMMs run
        {
            long nt = (long)tile + gridDim.x;
            if (nt < kNT)
                async_tile_prefetch(X, J, lds, buf ^ 1, nt * kTS, tid);
        }

        // ================= GEMM 1: Y = Wa@a + Wb@b + Wc@c =================
        v8f yacc[3] = {v8f{}, v8f{}, v8f{}};
        #pragma unroll
        for (int t = 0; t < 3; ++t) {
            const float* wl = &lds[W_OFF + t * W_SZ + arow * PAD];
            #pragma unroll
            for (int k = 0; k < 16; ++k) {
                int kk = 4 * k + 2 * lhalf;
                v2f A = *(const v2f*)(wl + kk);
                #pragma unroll
                for (int i = 0; i < 3; ++i) {
                    v2f Bv = *(const v2f*)(&lds[TERM_IDX(t, i) + lmod * PAD + kk]);
                    yacc[i] = __builtin_amdgcn_wmma_f32_16x16x4_f32(
                        false, A, false, Bv, (short)0, yacc[i], false, false);
                }
            }
        }

        // stage Y into LDS as [i][site][g] for the second contraction (over g=f)
        #pragma unroll
        for (int i = 0; i < 3; ++i)
            #pragma unroll
            for (int r = 0; r < 8; ++r) {
                int g = wave * 16 + r + 8 * lhalf;      // D-layout row
                lds[Y_OFF + i * T_SZ + lmod * PAD + g] = yacc[i][r];
            }
        __syncthreads();                               // barrier 2: Y staged

        // ================= GEMM 2: d = Wd @ Y =================
        v8f dacc[3] = {v8f{}, v8f{}, v8f{}};
        {
            const float* wl = &lds[W_OFF + 3 * W_SZ + arow * PAD];
            #pragma unroll
            for (int k = 0; k < 16; ++k) {
                int kk = 4 * k + 2 * lhalf;
                v2f A = *(const v2f*)(wl + kk);
                #pragma unroll
                for (int i = 0; i < 3; ++i) {
                    v2f Bv = *(const v2f*)(&lds[Y_OFF + i * T_SZ + lmod * PAD + kk]);
                    dacc[i] = __builtin_amdgcn_wmma_f32_16x16x4_f32(
                        false, A, false, Bv, (short)0, dacc[i], false, false);
                }
            }
        }

        // ========== epilogue: VN leaky relu, fully per-lane ==========
        long site = site0 + lmod;
        long bidx = site >> 14;          // / 16384
        long cidx = site & 16383;
        #pragma unroll
        for (int r = 0; r < 8; ++r) {
            int f = wave * 16 + r + 8 * lhalf;
            float y0 = yacc[0][r], y1 = yacc[1][r], y2 = yacc[2][r];
            float d0 = dacc[0][r], d1 = dacc[1][r], d2 = dacc[2][r];
            float dot = y0 * d0 + y1 * d1 + y2 * d2;
            float dn  = d0 * d0 + d1 * d1 + d2 * d2;
            float sc  = (dot >= 0.f) ? 0.f : dot / (dn + EPSF);
            float o0 = NEGS * y0 + (1.f - NEGS) * (y0 - sc * d0);
            float o1 = NEGS * y1 + (1.f - NEGS) * (y1 - sc * d1);
            float o2 = NEGS * y2 + (1.f - NEGS) * (y2 - sc * d2);
            long obase = ((bidx * 64 + f) * 3) * 16384 + cidx;
            out[obase]             = o0;
            out[obase + 16384]     = o1;
            out[obase + 2 * 16384] = o2;
        }

        // retire the async prefetch, then publish buffers for the next tile
        asm volatile("s_wait_asynccnt 0x0" ::: "memory");
        __syncthreads();                               // barrier 3
        buf ^= 1;
    }
}

extern "C" void kernel_launch(void* const* d_in, const int* in_sizes, int n_in,
                              void* d_out, int out_size, void* d_ws, size_t ws_size,
                              hipStream_t stream) {
    const float* X  = (const float*)d_in[0];
    const float* J  = (const float*)d_in[1];
    const float* Wa = (const float*)d_in[2];
    const float* Wb = (const float*)d_in[3];
    const float* Wc = (const float*)d_in[4];
    const float* Wd = (const float*)d_in[5];
    float* out = (float*)d_out;

    const size_t shmem = (size_t)LDS_FLOATS * sizeof(float);   // ~163.5 KB
    (void)hipFuncSetAttribute((const void*)vn_complex_fused_kernel,
                              hipFuncAttributeMaxDynamicSharedMemorySize, (int)shmem);
    vn_complex_fused_kernel<<<dim3(1024), dim3(128), shmem, stream>>>(
        X, J, Wa, Wb, Wc, Wd, out);
}